// AttentionWithKVCache_45861660787374
// MI455X (gfx1250) — compile-verified
//
#include <hip/hip_runtime.h>
#include <hip/hip_bf16.h>
#include <math.h>

// ---------------- problem constants (from the reference) ----------------
#define BATCH   16
#define DIM     4096
#define NHEADS  32
#define NKV     8
#define REP     4          // NHEADS / NKV
#define HD      128
#define PREFIX  8192
#define NSPLIT  16         // flash-decode wave-splits per (b, kv-head)
#define KEYS_PER_SPLIT (PREFIX / NSPLIT)   // 512

typedef __attribute__((ext_vector_type(2))) float v2f;
typedef __attribute__((ext_vector_type(4))) float v4f;
typedef __attribute__((ext_vector_type(8))) float v8f;

// ---------------------------------------------------------------------------
// Kernel 1/4: Y[16, Ncols] = X[16, K] * W[Ncols, K]^T   (fp32 WMMA 16x16x4)
// One wave handles one 16-column tile; M=16 == batch exactly fills the tile.
// W is streamed with non-temporal hints (no reuse; keep L2 for hot data).
//
// ISA VGPR layout (CDNA5 ISA):
//   A 16x4 : lanes 0-15 -> M=lane, {K0,K1}; lanes 16-31 -> M=lane-16, {K2,K3}
//   B 4x16 : lanes 0-15 -> N=lane, rows {K0,K1}; lanes 16-31 -> rows {K2,K3}
//   D 16x16: VGPR v, lanes 0-15 -> M=v; lanes 16-31 -> M=v+8; N=lane&15
// ---------------------------------------------------------------------------
__global__ __launch_bounds__(256)
void gemm16_wmma(const float* __restrict__ X, const float* __restrict__ W,
                 float* __restrict__ Y, int K, int Ncols) {
    const int wave = threadIdx.x >> 5;
    const int lane = threadIdx.x & 31;
    const int tile = blockIdx.x * 8 + wave;       // wave-uniform
    if (tile * 16 >= Ncols) return;               // whole wave exits together

    const int rc    = lane & 15;                  // M for A, N for B
    const int khalf = (lane >> 4) * 2;            // lane half selects K pair

    const float* xp = X + rc * K + khalf;
    const float* wp = W + (tile * 16 + rc) * (long)K + khalf;

    v8f c = {};
#pragma unroll 8
    for (int k0 = 0; k0 < K; k0 += 4) {
        v2f a = *(const v2f*)(xp + k0);
        v2f b = __builtin_nontemporal_load((const v2f*)(wp + k0));
        // 8 args: (neg_a, A, neg_b, B, c_mod, C, reuse_a, reuse_b)
        c = __builtin_amdgcn_wmma_f32_16x16x4_f32(false, a, false, b,
                                                  (short)0, c, false, false);
    }

    const int mbase = (lane >> 4) * 8;
    const int n     = tile * 16 + (lane & 15);
#pragma unroll
    for (int v = 0; v < 8; ++v)
        Y[(mbase + v) * (long)Ncols + n] = c[v];
}

// ---------------------------------------------------------------------------
// Kernel 2/4: flash-decode partial pass.
// One wave per (b, kvh, split): streams 512 K/V rows (float4/lane, fully
// coalesced 512B rows, non-temporal), online softmax for the REP=4 query
// heads sharing this kv head.  The running max is wave-uniform, so the
// accumulator rescale sits behind a uniform, rarely-taken branch.
// ---------------------------------------------------------------------------
__device__ __forceinline__ float waveSum32(float v) {
#pragma unroll
    for (int off = 16; off > 0; off >>= 1) v += __shfl_xor(v, off, 32);
    return v;
}

__device__ __forceinline__ void online_update(const float4 q4[REP],
                                              const float4& kv, const float4& vv,
                                              float m[REP], float l[REP],
                                              float4 acc[REP], float scale) {
#pragma unroll
    for (int j = 0; j < REP; ++j) {
        float d = q4[j].x * kv.x + q4[j].y * kv.y +
                  q4[j].z * kv.z + q4[j].w * kv.w;
        d = waveSum32(d) * scale;                 // wave-uniform result
        if (d > m[j]) {                           // uniform branch, rare
            const float alpha = __expf(m[j] - d);
            l[j] *= alpha;
            acc[j].x *= alpha; acc[j].y *= alpha;
            acc[j].z *= alpha; acc[j].w *= alpha;
            m[j] = d;
        }
        const float p = __expf(d - m[j]);
        l[j] += p;
        acc[j].x += p * vv.x; acc[j].y += p * vv.y;
        acc[j].z += p * vv.z; acc[j].w += p * vv.w;
    }
}

#define PART_STRIDE 132   // floats per (head,split) entry: m,l,pad,pad,acc[128]

__global__ __launch_bounds__(256)
void attn_partial(const float* __restrict__ Q,
                  const float* __restrict__ Knew,
                  const float* __restrict__ Vnew,
                  const float* __restrict__ cache_k,
                  const float* __restrict__ cache_v,
                  float* __restrict__ partials) {
    const int wave = threadIdx.x >> 5;
    const int lane = threadIdx.x & 31;
    const int gw   = blockIdx.x * 8 + wave;             // 0 .. 2047
    const int ws   = gw % NSPLIT;
    const int kvh  = (gw / NSPLIT) % NKV;
    const int b    = gw / (NSPLIT * NKV);

    const float scale = 0.08838834764831845f;           // 1/sqrt(128)

    // preload the 4 query heads of this GQA group (4 dims per lane)
    float4 q4[REP];
#pragma unroll
    for (int j = 0; j < REP; ++j)
        q4[j] = *(const float4*)(Q + b * DIM + (kvh * REP + j) * HD + 4 * lane);

    float  m[REP], l[REP];
    float4 acc[REP];
#pragma unroll
    for (int j = 0; j < REP; ++j) {
        m[j] = -INFINITY; l[j] = 0.f;
        acc[j] = make_float4(0.f, 0.f, 0.f, 0.f);
    }

    const long rowStride = (long)NKV * HD;              // floats per token
    const float* kp = cache_k + ((long)b * PREFIX * NKV + kvh) * HD + 4 * lane;
    const float* vp = cache_v + ((long)b * PREFIX * NKV + kvh) * HD + 4 * lane;

    const int kstart = ws * KEYS_PER_SPLIT;
    const int kend   = kstart + KEYS_PER_SPLIT;

#pragma unroll 4
    for (int k = kstart; k < kend; ++k) {
        const float* kr = kp + k * rowStride;
        const float* vr = vp + k * rowStride;
        // gfx1250 global_prefetch_b8 on the streams, ~8 keys ahead
        __builtin_prefetch(kr + 8 * rowStride, 0, 1);
        __builtin_prefetch(vr + 8 * rowStride, 0, 1);
        const v4f kv4 = __builtin_nontemporal_load((const v4f*)kr);
        const v4f vv4 = __builtin_nontemporal_load((const v4f*)vr);
        const float4 kv = make_float4(kv4.x, kv4.y, kv4.z, kv4.w);
        const float4 vv = make_float4(vv4.x, vv4.y, vv4.z, vv4.w);
        online_update(q4, kv, vv, m, l, acc, scale);
    }

    // the freshly projected token (cache "concat" position PREFIX)
    if (ws == NSPLIT - 1) {
        const float4 kv = *(const float4*)(Knew + b * (NKV * HD) + kvh * HD + 4 * lane);
        const float4 vv = *(const float4*)(Vnew + b * (NKV * HD) + kvh * HD + 4 * lane);
        online_update(q4, kv, vv, m, l, acc, scale);
    }

#pragma unroll
    for (int j = 0; j < REP; ++j) {
        float* base = partials +
            ((((long)b * NKV + kvh) * NSPLIT + ws) * REP + j) * PART_STRIDE;
        if (lane == 0) { base[0] = m[j]; base[1] = l[j]; }
        *(float4*)(base + 4 + 4 * lane) = acc[j];
    }
}

// ---------------------------------------------------------------------------
// Kernel 3/4: combine the NSPLIT partials per (b, head) into O[b, h, d].
// ---------------------------------------------------------------------------
__global__ __launch_bounds__(128)
void attn_combine(const float* __restrict__ partials, float* __restrict__ O) {
    const int bh  = blockIdx.x;                 // 0 .. B*NHEADS-1
    const int b   = bh / NHEADS;
    const int h   = bh % NHEADS;
    const int kvh = h / REP;
    const int j   = h % REP;
    const int d   = threadIdx.x;                // 0 .. 127

    const float* base0 = partials +
        (((long)b * NKV + kvh) * NSPLIT * REP + j) * PART_STRIDE;

    float M = -INFINITY;
#pragma unroll
    for (int ws = 0; ws < NSPLIT; ++ws)
        M = fmaxf(M, base0[(long)ws * REP * PART_STRIDE]);

    float L = 0.f, val = 0.f;
#pragma unroll
    for (int ws = 0; ws < NSPLIT; ++ws) {
        const float* e = base0 + (long)ws * REP * PART_STRIDE;
        const float w = __expf(e[0] - M);
        L   += e[1] * w;
        val += e[4 + d] * w;
    }
    O[(long)b * DIM + h * HD + d] = val / L;
}

// ---------------------------------------------------------------------------
// launch
// ---------------------------------------------------------------------------
extern "C" void kernel_launch(void* const* d_in, const int* in_sizes, int n_in,
                              void* d_out, int out_size, void* d_ws, size_t ws_size,
                              hipStream_t stream) {
    const float* x       = (const float*)d_in[0];   // [16,1,4096]
    const float* cache_k = (const float*)d_in[1];   // [16,8192,8,128]
    const float* cache_v = (const float*)d_in[2];
    const float* wq      = (const float*)d_in[3];   // [4096,4096]
    const float* wk      = (const float*)d_in[4];   // [1024,4096]
    const float* wv      = (const float*)d_in[5];   // [1024,4096]
    const float* wo      = (const float*)d_in[6];   // [4096,4096]
    float*       out     = (float*)d_out;           // [16,1,4096]

    // workspace layout (floats)
    float* ws       = (float*)d_ws;
    float* Q        = ws;                         // 16*4096
    float* Knew     = Q + BATCH * DIM;            // 16*1024
    float* Vnew     = Knew + BATCH * NKV * HD;    // 16*1024
    float* O        = Vnew + BATCH * NKV * HD;    // 16*4096
    float* partials = O + BATCH * DIM;            // 16*8*16*4*132

    // 1) QKV projections (fp32 WMMA, M=16)
    gemm16_wmma<<<(NHEADS * HD) / (16 * 8), 256, 0, stream>>>(x, wq, Q,    DIM, NHEADS * HD);
    gemm16_wmma<<<(NKV    * HD) / (16 * 8), 256, 0, stream>>>(x, wk, Knew, DIM, NKV    * HD);
    gemm16_wmma<<<(NKV    * HD) / (16 * 8), 256, 0, stream>>>(x, wv, Vnew, DIM, NKV    * HD);

    // 2) flash-decode partial pass: 2048 waves, 8 waves/block
    attn_partial<<<(BATCH * NKV * NSPLIT) / 8, 256, 0, stream>>>(
        Q, Knew, Vnew, cache_k, cache_v, partials);

    // 3) combine partials
    attn_combine<<<BATCH * NHEADS, 128, 0, stream>>>(partials, O);

    // 4) output projection (fp32 WMMA)
    gemm16_wmma<<<DIM / (16 * 8), 256, 0, stream>>>(O, wo, out, DIM, DIM);
}